// AdaptivePruner_36558761624183
// MI455X (gfx1250) — compile-verified
//
#include <hip/hip_runtime.h>
#include <hip/hip_bf16.h>

typedef __attribute__((ext_vector_type(2))) float v2f;
typedef __attribute__((ext_vector_type(8))) float v8f;

// Toggle for the gfx1250 async global->LDS copy path (inline asm, ISA §15.18.3).
#define USE_ASYNC_COPY 1

// db4 dec_lo reversed (cross-correlation form): y[k] = sum_j h[j] * x[2k + j - 6]
__device__ __constant__ float kH[8] = {
     0.23037781330885523f,  0.7148465705525415f,  0.6308807679295904f,
    -0.02798376941698385f, -0.18703481171888114f, 0.030841381835986965f,
     0.032883011666982945f, -0.010597401784997278f
};

// ---------------- Kernel 1: per-batch entropy -------------------------------
__global__ __launch_bounds__(256) void entropy_kernel(const float* __restrict__ att,
                                                      float* __restrict__ ent) {
    const int b = blockIdx.x;
    const int tid = threadIdx.x;
    __shared__ float red[256];
    float v = 0.0f;
    if (tid < 196) {
        float p = att[b * 196 + tid];
        v = p * log2f(p + 1e-9f);
    }
    red[tid] = v;
    __syncthreads();
    for (int s = 128; s > 0; s >>= 1) {
        if (tid < s) red[tid] += red[tid + s];
        __syncthreads();
    }
    if (tid == 0) ent[b] = -red[0];
}

// ---------------- Kernel 2: quantiles, levels, attention mask ---------------
__global__ __launch_bounds__(256) void quantile_mask_kernel(const float* __restrict__ ent,
                                                            int* __restrict__ levels,
                                                            float* __restrict__ maskout) {
    const int tid = threadIdx.x;
    __shared__ float s[256];
    __shared__ float q[2];
    s[tid] = ent[tid];
    __syncthreads();
    // bitonic sort (ascending) of 256 values
    for (int k = 2; k <= 256; k <<= 1) {
        for (int j = k >> 1; j > 0; j >>= 1) {
            int ixj = tid ^ j;
            if (ixj > tid) {
                float a = s[tid], c = s[ixj];
                bool up = ((tid & k) == 0);
                if ((a > c) == up) { s[tid] = c; s[ixj] = a; }
            }
            __syncthreads();
        }
    }
    if (tid == 0) {
        // jnp.quantile linear interpolation: pos = q*(n-1); n=256
        q[0] = s[63] + 0.75f * (s[64] - s[63]);   // q = 0.25 -> pos 63.75
        q[1] = 0.5f * (s[127] + s[128]);          // q = 0.50 -> pos 127.5
    }
    __syncthreads();
    float e = ent[tid];
    int lev = 3 - (int)(e > q[0]) - (int)(e > q[1]);
    levels[tid] = lev;
    int out_len = (lev == 1) ? 101 : ((lev == 2) ? 54 : 30);
    float* row = maskout + tid * 102;
    row[0] = 1.0f;
    for (int i = 0; i < 101; ++i) row[1 + i] = (i < out_len) ? 1.0f : 0.0f;
}

// ---------------- WMMA banded-matrix DWT level (LDS -> LDS) -----------------
// Buffers are zero-padded: row index = (t + 8); reads up to t = 2*m0+33 and
// unguarded writes up to m = m0+15 stay inside the padded region, and the
// out-of-range conv outputs of a zero-extended signal are exactly zero.
__device__ __forceinline__ void dwt_level(const float* __restrict__ Yin,
                                          float* __restrict__ Yout,
                                          int Mout, int la, int hi,
                                          const v2f (&afrag)[10]) {
    for (int m0 = 0; m0 < Mout; m0 += 16) {
        v8f acc = {};
        // half-wave K split: lanes 16-31 take K+2  =>  t = 2*m0 - 6 + 2*hi + 4*kk (+1)
        const float* in = Yin + (2 * m0 - 6 + 2 * hi + 8) * 16 + la;
        #pragma unroll
        for (int kk = 0; kk < 10; ++kk) {
            v2f bfrag;
            bfrag.x = in[kk * 64];        // ds_load with immediate offset kk*256B
            bfrag.y = in[kk * 64 + 16];   // next row (+64B)
            acc = __builtin_amdgcn_wmma_f32_16x16x4_f32(
                /*neg_a=*/false, afrag[kk], /*neg_b=*/false, bfrag,
                /*c_mod=*/(short)0, acc, /*reuse_a=*/false, /*reuse_b=*/false);
        }
        // C/D layout: VGPR r holds M=m0+r (lanes 0-15), M=m0+r+8 (lanes 16-31)
        float* o = Yout + (m0 + 8 * hi + 8) * 16 + la;
        #pragma unroll
        for (int r = 0; r < 8; ++r) o[r * 16] = acc[r];
    }
}

// ---------------- Kernel 3: 3-level DWT + per-batch selection ---------------
// grid = (48 feature tiles of 16, 256 batches), block = one wave32
__global__ __launch_bounds__(32) void dwt_select_kernel(const float* __restrict__ x,
                                                        const int* __restrict__ levels,
                                                        float* __restrict__ out) {
    constexpr int RX = 236;   // 8 front pad + 196 valid + back pad (reads to t=225)
    constexpr int R1 = 140;   // 8 + writes to m=111, reads to t=129
    constexpr int R2 = 76;    // 8 + writes to m=63,  reads to t=65
    constexpr int R3 = 40;    // 8 + writes to m=31
    constexpr int RTOT = RX + R1 + R2 + R3;   // 492 rows * 16 f32 = 30.75 KB

    const int d0   = blockIdx.x * 16;
    const int b    = blockIdx.y;
    const int lane = threadIdx.x;
    const int la   = lane & 15;
    const int hi   = lane >> 4;

    __shared__ float hsh[8];
    __shared__ float BUF[RTOT * 16];
    float* Xls = BUF;
    float* Y1  = BUF + RX * 16;
    float* Y2  = Y1 + R1 * 16;
    float* Y3  = Y2 + R2 * 16;

    // zero-fill all stage buffers (b128 stores) + filter taps
    {
        float4  z4 = make_float4(0.0f, 0.0f, 0.0f, 0.0f);
        float4* z  = (float4*)BUF;
        for (int i = lane; i < RTOT * 4; i += 32) z[i] = z4;
        if (lane < 8) hsh[lane] = kH[lane];
    }
    __syncthreads();   // DScnt drained: zeros committed before async writes land

    const float* xb = x + (size_t)b * 197 * 768;
    float* ob = out + (size_t)b * 102 * 768;

    // ---- stage patch-token tile (tokens 1..196, features d0..d0+15) --------
#if USE_ASYNC_COPY
    for (int idx = lane; idx < 196 * 4; idx += 32) {
        int t = idx >> 2, q = idx & 3;
        const float* g = xb + (size_t)(1 + t) * 768 + d0 + 4 * q;
        unsigned ldsaddr = (unsigned)(size_t)(Xls + (8 + t) * 16 + 4 * q);
        unsigned long long ga = (unsigned long long)(size_t)g;
        asm volatile("global_load_async_to_lds_b128 %0, %1, off"
                     :: "v"(ldsaddr), "v"(ga) : "memory");
    }
#else
    for (int idx = lane; idx < 196 * 4; idx += 32) {
        int t = idx >> 2, q = idx & 3;
        *(float4*)(Xls + (8 + t) * 16 + 4 * q) =
            *(const float4*)(xb + (size_t)(1 + t) * 768 + d0 + 4 * q);
    }
#endif

    // cls-token passthrough (overlaps the staging copy)
    if (lane < 4) *(float4*)(ob + d0 + 4 * lane) = *(const float4*)(xb + d0 + 4 * lane);

    // A-matrix fragments: W[m, t] = h[t - (2m - 6)] => j = 4*kk + (2*hi + e) - 2*la,
    // independent of tile and level. (hsh reads overlap the async copy: disjoint LDS.)
    v2f afrag[10];
    #pragma unroll
    for (int kk = 0; kk < 10; ++kk) {
        int j0 = 4 * kk + 2 * hi - 2 * la;
        int j1 = j0 + 1;
        int c0 = (j0 < 0) ? 0 : ((j0 > 7) ? 7 : j0);
        int c1 = (j1 < 0) ? 0 : ((j1 > 7) ? 7 : j1);
        float h0 = hsh[c0], h1 = hsh[c1];
        afrag[kk].x = (j0 >= 0 && j0 < 8) ? h0 : 0.0f;
        afrag[kk].y = (j1 >= 0 && j1 < 8) ? h1 : 0.0f;
    }

#if USE_ASYNC_COPY
    asm volatile("s_wait_asynccnt 0" ::: "memory");
#endif
    __syncthreads();

    dwt_level(Xls, Y1, 101, la, hi, afrag);
    __syncthreads();
    dwt_level(Y1, Y2, 54, la, hi, afrag);
    __syncthreads();
    dwt_level(Y2, Y3, 30, la, hi, afrag);
    __syncthreads();

    // per-batch level selection; write full 101-token block (zeros in tail)
    const int lev = levels[b];
    const float4 z4 = make_float4(0.0f, 0.0f, 0.0f, 0.0f);
    for (int idx = lane; idx < 101 * 4; idx += 32) {
        int k = idx >> 2, q = idx & 3;
        float4 v;
        if (lev == 1)      v = *((const float4*)(Y1 + (k + 8) * 16) + q);
        else if (lev == 2) v = (k < 54) ? *((const float4*)(Y2 + (k + 8) * 16) + q) : z4;
        else               v = (k < 30) ? *((const float4*)(Y3 + (k + 8) * 16) + q) : z4;
        *(float4*)(ob + (size_t)(1 + k) * 768 + d0 + 4 * q) = v;
    }
}

extern "C" void kernel_launch(void* const* d_in, const int* in_sizes, int n_in,
                              void* d_out, int out_size, void* d_ws, size_t ws_size,
                              hipStream_t stream) {
    (void)in_sizes; (void)n_in; (void)out_size; (void)ws_size;
    const float* x   = (const float*)d_in[0];   // (256, 197, 768) f32
    const float* att = (const float*)d_in[1];   // (256, 196) f32
    float* out = (float*)d_out;                 // final_x (256,102,768) ++ mask (256,102)

    float* ent     = (float*)d_ws;
    int*   levels  = (int*)((char*)d_ws + 256 * sizeof(float));
    float* maskout = out + (size_t)256 * 102 * 768;

    entropy_kernel<<<256, 256, 0, stream>>>(att, ent);
    quantile_mask_kernel<<<1, 256, 0, stream>>>(ent, levels, maskout);
    dwt_select_kernel<<<dim3(48, 256), 32, 0, stream>>>(x, levels, out);
}